// LSTM_45638322487333
// MI455X (gfx1250) — compile-verified
//
#include <hip/hip_runtime.h>

// ---------------- problem constants ----------------
#define NWG   32
#define TSEQ  2048
#define BB    64
#define DD    64
#define HH    512
#define OO    128

typedef __attribute__((ext_vector_type(2))) float v2f;
typedef __attribute__((ext_vector_type(8))) float v8f;

// ---------------- LDS layout (dynamic) ----------------
#define WHH_STRIDE   516                      // 516 % 64 == 4 -> conflict-free f2 reads
#define WHH_FLOATS   (64 * WHH_STRIDE)        // 33024
#define G_STRIDE     17                       // per-batch pad for gate exchange
#define G_PLANE      (64 * G_STRIDE)          // 1088 floats per gate type
#define GATES_FLOATS (4 * G_PLANE)            // 4352
#define CLDS_FLOATS  (16 * 64)                // c state: [n][b]
#define BIAS_FLOATS  64
#define SMEM_FLOATS  (WHH_FLOATS + GATES_FLOATS + CLDS_FLOATS + BIAS_FLOATS) // 38464 -> 153856 B

__device__ __forceinline__ float sigm(float x)  { return 1.0f / (1.0f + __expf(-x)); }
__device__ __forceinline__ float tanhx(float x) { return 1.0f - 2.0f / (__expf(2.0f * x) + 1.0f); }

__global__ void lstm_init_kernel(float* __restrict__ hbuf, unsigned* __restrict__ cnt) {
    int i = blockIdx.x * blockDim.x + threadIdx.x;
    if (i < HH * BB) hbuf[i] = 0.0f;          // h0 = 0 (buffer 0, transposed [H][B])
    if (i == 0) *cnt = 0u;
}

__global__ void __launch_bounds__(512, 1)
lstm_persistent(const float* __restrict__ x,     // [B][T][D]
                const float* __restrict__ W_ih,  // [4H][D]
                const float* __restrict__ W_hh,  // [4H][H]
                const float* __restrict__ b_ih,  // [4H]
                const float* __restrict__ b_hh,  // [4H]
                const float* __restrict__ W_out, // [O][H]
                const float* __restrict__ b_out, // [O]
                float* __restrict__ out,         // pred(8192) | h_n(32768) | c_n(32768)
                float* __restrict__ hbuf,        // [2][H][B] transposed, double buffer
                unsigned* __restrict__ cnt)      // grid-sync counter
{
    extern __shared__ float smem[];
    float* w_hh_l = smem;                      // [64][516]  this WG's 64 gate rows
    float* gates  = smem + WHH_FLOATS;         // [4][64][17] gate exchange
    float* c_lds  = gates + GATES_FLOATS;      // [16][64]   cell state (n-major)
    float* bias_l = c_lds + CLDS_FLOATS;       // [64]       b_ih + b_hh for our rows

    const int tid  = threadIdx.x;
    const int wg   = blockIdx.x;               // 0..31
    const int hu0  = wg * 16;                  // hidden-unit slice base
    const int lane = tid & 31;                 // wave32
    const int wv   = tid >> 5;                 // wave 0..15
    const int mt   = wv & 3;                   // batch tile
    const int gt   = wv >> 2;                  // gate type (i,f,g,o)
    const int b0   = mt * 16;
    const int n    = lane & 15;                // A row within tile / B column
    const int s    = lane >> 4;                // K half-select per WMMA layout

    // ---- stage W_hh slice (our 64 gate rows) + bias into LDS, zero c ----
    for (int i = 0; i < 64; ++i) {
        int idx = tid + 512 * i;               // 0..32767
        int r   = idx >> 9;                    // local gate row 0..63
        int k   = idx & 511;
        int rg  = (r >> 4) * HH + hu0 + (r & 15);
        w_hh_l[r * WHH_STRIDE + k] = W_hh[(size_t)rg * HH + k];
    }
    if (tid < 64) {
        int rg = (tid >> 4) * HH + hu0 + (tid & 15);
        bias_l[tid] = b_ih[rg] + b_hh[rg];
    }
    for (int p = tid; p < CLDS_FLOATS; p += 512) c_lds[p] = 0.0f;
    __syncthreads();

    const float* wih_lane = W_ih + (size_t)(gt * HH + hu0 + n) * DD; // B rows for x-proj
    const float* whh_lane = w_hh_l + (gt * 16 + n) * WHH_STRIDE;     // B rows for h-GEMM

    unsigned target = 0;
    for (int t = 0; t < TSEQ; ++t) {
        const float* hread  = hbuf + ((t)     & 1) * (HH * BB);
        float*       hwrite = hbuf + ((t + 1) & 1) * (HH * BB);

        // acc = bias (broadcast down the column)
        float bv = bias_l[gt * 16 + n];
        v8f acc = { bv, bv, bv, bv, bv, bv, bv, bv };

        // ---- fused input projection: G += x_t[16b x 64k] * W_ih^T ----
        const float* x_lane = x + ((size_t)(b0 + n) * TSEQ + t) * DD; // A row = batch
#pragma unroll
        for (int kk = 0; kk < 16; ++kk) {
            int k0 = kk * 4 + 2 * s;
            v2f a = *(const v2f*)(x_lane + k0);
            v2f b = *(const v2f*)(wih_lane + k0);
            acc = __builtin_amdgcn_wmma_f32_16x16x4_f32(false, a, false, b,
                                                        (short)0, acc, false, false);
        }
        // ---- recurrent GEMM: G += h[16b x 512k] * W_hh^T (LDS-resident) ----
#pragma unroll 4
        for (int kk = 0; kk < 128; ++kk) {
            int k0 = kk * 4 + 2 * s;
            v2f a;
            a.x = hread[(k0    ) * BB + b0 + n];   // h transposed [H][B]: coalesced
            a.y = hread[(k0 + 1) * BB + b0 + n];
            v2f b = *(const v2f*)(whh_lane + k0);
            acc = __builtin_amdgcn_wmma_f32_16x16x4_f32(false, a, false, b,
                                                        (short)0, acc, false, false);
        }

        // ---- exchange 16x16 gate tiles through LDS ----
#pragma unroll
        for (int v = 0; v < 8; ++v) {
            int m = v + 8 * s;                  // C layout: M = vgpr + 8*(lane>=16)
            gates[gt * G_PLANE + (b0 + m) * G_STRIDE + n] = acc[v];
        }
        __syncthreads();

        // ---- elementwise LSTM cell update; write new h (transposed, coalesced) ----
        for (int p = tid; p < 1024; p += 512) {
            int nn = p >> 6;                    // hidden local 0..15
            int bb = p & 63;                    // batch
            int gi = bb * G_STRIDE + nn;
            float iv = sigm (gates[0 * G_PLANE + gi]);
            float fv = sigm (gates[1 * G_PLANE + gi]);
            float gv = tanhx(gates[2 * G_PLANE + gi]);
            float ov = sigm (gates[3 * G_PLANE + gi]);
            float cn = fv * c_lds[nn * 64 + bb] + iv * gv;
            c_lds[nn * 64 + bb] = cn;
            hwrite[(hu0 + nn) * BB + bb] = ov * tanhx(cn);
        }

        // ---- grid-wide sync (release writes, acquire others') ----
        target += NWG;
        __syncthreads();
        if (tid == 0) {
            __hip_atomic_fetch_add(cnt, 1u, __ATOMIC_RELEASE, __HIP_MEMORY_SCOPE_AGENT);
            while (__hip_atomic_load(cnt, __ATOMIC_RELAXED, __HIP_MEMORY_SCOPE_AGENT) < target)
                __builtin_amdgcn_s_sleep(2);
        }
        __syncthreads();
        __builtin_amdgcn_fence(__ATOMIC_ACQUIRE, "agent");
    }

    // T is even -> final h lives in buffer 0 (visible after last grid sync)
    const float* hfin = hbuf;

    // ---- emit h_n and c_n for our slice ----
    for (int p = tid; p < 1024; p += 512) {
        int nn = p >> 6, bb = p & 63;
        out[8192  + (size_t)bb * HH + hu0 + nn] = hfin[(hu0 + nn) * BB + bb];
        out[40960 + (size_t)bb * HH + hu0 + nn] = c_lds[nn * 64 + bb];
    }

    // ---- pred = hT @ W_out^T + b_out, done by WG 0 (32 tiles / 16 waves) ----
    if (wg == 0) {
        for (int j = 0; j < 2; ++j) {
            int tile = wv * 2 + j;
            int bt0 = (tile & 3) * 16;
            int n0  = (tile >> 2) * 16;
            float bo = b_out[n0 + n];
            v8f a2 = { bo, bo, bo, bo, bo, bo, bo, bo };
            const float* wob = W_out + (size_t)(n0 + n) * HH;
#pragma unroll 4
            for (int kk = 0; kk < 128; ++kk) {
                int k0 = kk * 4 + 2 * s;
                v2f a;
                a.x = hfin[(k0    ) * BB + bt0 + n];
                a.y = hfin[(k0 + 1) * BB + bt0 + n];
                v2f b = *(const v2f*)(wob + k0);
                a2 = __builtin_amdgcn_wmma_f32_16x16x4_f32(false, a, false, b,
                                                           (short)0, a2, false, false);
            }
#pragma unroll
            for (int v = 0; v < 8; ++v) {
                int m = v + 8 * s;
                out[(size_t)(bt0 + m) * OO + n0 + n] = a2[v];
            }
        }
    }
}

extern "C" void kernel_launch(void* const* d_in, const int* in_sizes, int n_in,
                              void* d_out, int out_size, void* d_ws, size_t ws_size,
                              hipStream_t stream) {
    const float* x     = (const float*)d_in[0];
    const float* W_ih  = (const float*)d_in[1];
    const float* W_hh  = (const float*)d_in[2];
    const float* b_ih  = (const float*)d_in[3];
    const float* b_hh  = (const float*)d_in[4];
    const float* W_out = (const float*)d_in[5];
    const float* b_out = (const float*)d_in[6];
    // d_in[7] = batch_size scalar (unused; B fixed at 64)
    float* out = (float*)d_out;

    float*    hbuf = (float*)d_ws;                                   // 2*512*64 f32 = 256 KB
    unsigned* cnt  = (unsigned*)((char*)d_ws + 2u * HH * BB * sizeof(float));

    lstm_init_kernel<<<(HH * BB + 255) / 256, 256, 0, stream>>>(hbuf, cnt);
    lstm_persistent<<<NWG, 512, SMEM_FLOATS * sizeof(float), stream>>>(
        x, W_ih, W_hh, b_ih, b_hh, W_out, b_out, out, hbuf, cnt);
}